// SAGE_5016521801890
// MI455X (gfx1250) — compile-verified
//
#include <hip/hip_runtime.h>
#include <hip/hip_bf16.h>
#include <stdint.h>

#define N_NODES 50000
#define N_EDGES 800000
#define IN_F    100
#define N_HID   256
#define N_CLS   47

typedef float v2f __attribute__((ext_vector_type(2)));
typedef float v8f __attribute__((ext_vector_type(8)));

#if __has_builtin(__builtin_amdgcn_wmma_f32_16x16x4_f32)
#define USE_WMMA_F32 1
#endif

// One wave computes a 16x16 f32 tile, K-step 4 (V_WMMA_F32_16X16X4_F32).
// A frag (16x4, 32-bit): lane L<16 holds M=L, K={0,1}; lane L>=16 holds M=L-16, K={2,3}.
// B frag (4x16, 32-bit): lane L holds N=L%16, K={0,1} (L<16) or K={2,3} (L>=16).
__device__ __forceinline__ v8f wmma4(v2f a, v2f b, v8f c) {
#ifdef USE_WMMA_F32
  return __builtin_amdgcn_wmma_f32_16x16x4_f32(false, a, false, b, (short)0, c,
                                               false, false);
#else
  c[0] += a[0] * b[0] + a[1] * b[1];  // compile insurance only
  return c;
#endif
}

// LDS panel layout: element (k, n) lives at ((k>>1)*16 + n)*2 + (k&1), so the
// B fragment for even ka is one aligned 8-byte ds_load_b64 at ((ka>>1)*16+n)*2.
__device__ __forceinline__ v2f lds_bfrag(const float* p, int ka, int n) {
  return *reinterpret_cast<const v2f*>(p + (((ka >> 1) * 16 + n) << 1));
}

__global__ void zero_f32(float* __restrict__ p, int n) {
  int i = blockIdx.x * blockDim.x + threadIdx.x;
  int stride = gridDim.x * blockDim.x;
  for (; i < n; i += stride) p[i] = 0.0f;
}

__global__ void degree_k(const int* __restrict__ dst, float* __restrict__ deg) {
  int e = blockIdx.x * blockDim.x + threadIdx.x;
  if (e < N_EDGES) atomicAdd(&deg[dst[e]], 1.0f);
}

__global__ void invdeg_k(const float* __restrict__ deg, float* __restrict__ inv) {
  int i = blockIdx.x * blockDim.x + threadIdx.x;
  if (i < N_NODES) inv[i] = 1.0f / fmaxf(deg[i], 1.0f);
}

// Thread per (edge, float4 chunk): 800000 * 25 work items, all lanes active.
__global__ void scatter_x_k(const float* __restrict__ x, const int* __restrict__ src,
                            const int* __restrict__ dst, float* __restrict__ xagg) {
  const int idx = blockIdx.x * blockDim.x + threadIdx.x;  // exactly 20,000,000
  const int e = idx / 25;
  const int f = idx - e * 25;
  const int s = src[e], d = dst[e];
  if (f == 0 && e + 64 < N_EDGES)  // warm L2 for an upcoming gather row
    __builtin_prefetch(x + (size_t)src[e + 64] * IN_F, 0, 1);
  const float4 v = reinterpret_cast<const float4*>(x + (size_t)s * IN_F)[f];
  float* ar = xagg + (size_t)d * IN_F + f * 4;
  atomicAdd(&ar[0], v.x);
  atomicAdd(&ar[1], v.y);
  atomicAdd(&ar[2], v.z);
  atomicAdd(&ar[3], v.w);
}

__global__ void scale_k(float* __restrict__ xagg, const float* __restrict__ inv) {
  int i = blockIdx.x * blockDim.x + threadIdx.x;
  int stride = gridDim.x * blockDim.x;
  const int n = N_NODES * IN_F;
  for (; i < n; i += stride) xagg[i] *= inv[i / IN_F];
}

// h1d = dropout(relu(x @ Ws1 + xmean @ Wn1 + b1)).
// Block: fixed N-tile tn, 8 waves = 8 consecutive M-tiles. B panels staged in LDS.
__global__ void __launch_bounds__(256) gemm1_k(
    const float* __restrict__ x, const float* __restrict__ xm,
    const float* __restrict__ Ws, const float* __restrict__ Wn,
    const float* __restrict__ bias, float* __restrict__ out) {
  __shared__ float sBs[IN_F * 16];  // 6400 B
  __shared__ float sBn[IN_F * 16];
  const int tn = blockIdx.x & 15;         // 16 N-tiles
  const int mg = blockIdx.x >> 4;         // M-group of 8 tiles
  const int tid = threadIdx.x;
  // Stage both weight panels (cols tn*16..tn*16+15), interleaved layout.
  for (int i = tid; i < IN_F * 16; i += 256) {
    const int k = i >> 4, n = i & 15;
    const int pos = (((k >> 1) * 16 + n) << 1) + (k & 1);
    const int gcol = tn * 16 + n;
    sBs[pos] = Ws[(size_t)k * N_HID + gcol];
    sBn[pos] = Wn[(size_t)k * N_HID + gcol];
  }
  __syncthreads();

  const int wave = tid >> 5, lane = tid & 31;
  const int tm = mg * 8 + wave;
  if (tm >= N_NODES / 16) return;  // wave-uniform
  const int half = lane >> 4, l16 = lane & 15;
  const int row = tm * 16 + l16;
  const int col = tn * 16 + l16;
  const float* xr = x  + (size_t)row * IN_F;
  const float* mr = xm + (size_t)row * IN_F;
  v8f acc = {};
  for (int k0 = 0; k0 < IN_F; k0 += 4) {
    const int ka = k0 + half * 2;
    v2f aS = {xr[ka], xr[ka + 1]};
    v2f aN = {mr[ka], mr[ka + 1]};
    acc = wmma4(aS, lds_bfrag(sBs, ka, l16), acc);
    acc = wmma4(aN, lds_bfrag(sBn, ka, l16), acc);
  }
  const float bv = bias[col];
  for (int v = 0; v < 8; ++v) {
    const int m = tm * 16 + v + half * 8;
    float val = fmaxf(acc[v] + bv, 0.0f);
    uint32_t h = (uint32_t)(m * N_HID + col);
    h ^= h >> 16; h *= 0x7feb352du; h ^= h >> 15; h *= 0x846ca68bu; h ^= h >> 16;
    val = (h & 1u) ? val * 2.0f : 0.0f;  // p=0.5 keep, scale 1/(1-p)
    out[(size_t)m * N_HID + col] = val;
  }
}

// d_out = h1d @ Ws2 + b2 ; tneigh = h1d @ Wn2.
// Block: fixed N-tile tn (3 tiles cover 47 cols, zero-padded at staging),
// 8 waves = 8 consecutive M-tiles. B panels (256x16 each) staged in LDS.
__global__ void __launch_bounds__(256) gemm2_k(
    const float* __restrict__ h, const float* __restrict__ Ws,
    const float* __restrict__ Wn, const float* __restrict__ bias,
    float* __restrict__ out_self, float* __restrict__ tneigh) {
  __shared__ float sBs[N_HID * 16];  // 16 KB
  __shared__ float sBn[N_HID * 16];  // 16 KB
  const int NB = 3;
  const int tn = blockIdx.x % NB;
  const int mg = blockIdx.x / NB;
  const int tid = threadIdx.x;
  for (int i = tid; i < N_HID * 16; i += 256) {
    const int k = i >> 4, n = i & 15;
    const int pos = (((k >> 1) * 16 + n) << 1) + (k & 1);
    const int gcol = tn * 16 + n;
    const bool ok = gcol < N_CLS;
    sBs[pos] = ok ? Ws[(size_t)k * N_CLS + gcol] : 0.0f;
    sBn[pos] = ok ? Wn[(size_t)k * N_CLS + gcol] : 0.0f;
  }
  __syncthreads();

  const int wave = tid >> 5, lane = tid & 31;
  const int tm = mg * 8 + wave;
  if (tm >= N_NODES / 16) return;  // wave-uniform
  const int half = lane >> 4, l16 = lane & 15;
  const int col = tn * 16 + l16;
  const bool colOK = (col < N_CLS);
  const float* hr = h + (size_t)(tm * 16 + l16) * N_HID;
  v8f cs = {}, cn = {};
  for (int k0 = 0; k0 < N_HID; k0 += 4) {
    const int ka = k0 + half * 2;
    v2f a = {hr[ka], hr[ka + 1]};
    cs = wmma4(a, lds_bfrag(sBs, ka, l16), cs);
    cn = wmma4(a, lds_bfrag(sBn, ka, l16), cn);
  }
  if (colOK) {
    const float bv = bias[col];
    for (int v = 0; v < 8; ++v) {
      const int m = tm * 16 + v + half * 8;
      out_self[(size_t)m * N_CLS + col] = cs[v] + bv;
      tneigh[(size_t)m * N_CLS + col]   = cn[v];
    }
  }
}

// Thread per (edge, class): out[dst] += tneigh[src] * invdeg[dst].
__global__ void agg2_k(const int* __restrict__ src, const int* __restrict__ dst,
                       const float* __restrict__ tneigh, const float* __restrict__ inv,
                       float* __restrict__ out) {
  const int idx = blockIdx.x * blockDim.x + threadIdx.x;  // exactly 37,600,000
  const int e = idx / N_CLS;
  const int c = idx - e * N_CLS;
  const int s = src[e], d = dst[e];
  atomicAdd(&out[(size_t)d * N_CLS + c], tneigh[(size_t)s * N_CLS + c] * inv[d]);
}

extern "C" void kernel_launch(void* const* d_in, const int* in_sizes, int n_in,
                              void* d_out, int out_size, void* d_ws, size_t ws_size,
                              hipStream_t stream) {
  const float* x   = (const float*)d_in[0];
  const int*   src = (const int*)d_in[1];
  const int*   dst = (const int*)d_in[2];
  const float* Ws1 = (const float*)d_in[3];
  const float* Wn1 = (const float*)d_in[4];
  const float* b1  = (const float*)d_in[5];
  const float* Ws2 = (const float*)d_in[6];
  const float* Wn2 = (const float*)d_in[7];
  const float* b2  = (const float*)d_in[8];
  float* out = (float*)d_out;

  float* deg    = (float*)d_ws;                    // 50000
  float* invdeg = deg + N_NODES;                   // 50000
  float* xagg   = invdeg + N_NODES;                // 5,000,000
  float* h1     = xagg + (size_t)N_NODES * IN_F;   // 12,800,000
  float* tn     = h1 + (size_t)N_NODES * N_HID;    // 2,350,000   (~81 MB total)

  zero_f32<<<2048, 256, 0, stream>>>((float*)d_ws, N_NODES * (IN_F + 2));
  degree_k<<<(N_EDGES + 255) / 256, 256, 0, stream>>>(dst, deg);
  invdeg_k<<<(N_NODES + 255) / 256, 256, 0, stream>>>(deg, invdeg);
  scatter_x_k<<<(N_EDGES * 25) / 256, 256, 0, stream>>>(x, src, dst, xagg);
  scale_k<<<4096, 256, 0, stream>>>(xagg, invdeg);
  // GEMM1: 16 N-tiles x ceil(3125/8)=391 M-groups
  gemm1_k<<<16 * 391, 256, 0, stream>>>(x, xagg, Ws1, Wn1, b1, h1);
  // GEMM2: 3 N-tiles x 391 M-groups
  gemm2_k<<<3 * 391, 256, 0, stream>>>(h1, Ws2, Wn2, b2, out, tn);
  agg2_k<<<(N_EDGES * N_CLS) / 256, 256, 0, stream>>>(src, dst, tn, invdeg, out);
}